// NonMaximumSuppression_35381940584565
// MI455X (gfx1250) — compile-verified
//
#include <hip/hip_runtime.h>
#include <hip/hip_bf16.h>

// ---------------------------------------------------------------------------
// NMS for 5000 anchors x 80 classes on gfx1250 (MI455X).
//  K1: pad boxes to 5008, compute areas, zero mask, write out[:,0:4]
//  K2: 16x16 IoU tiles, 8 j-tiles per wave; union-sum term via
//      V_WMMA_F32_16X16X4_F32 (A=[area_i,1], B=[1;area_j]); the NMS predicate
//      is evaluated divide-free as inter > 0.5*max(union,1e-8) and packed to
//      bits with wave32 ballot -> 3.2MB suppression mask (L2-resident)
//  K3: one workgroup per class: LDS bitonic argsort (desc score, asc idx),
//      sequential scan with uint4 bitmask row-OR suppression, cap 300
// ---------------------------------------------------------------------------

#define N_ANCH   5000
#define N_PAD    5008            // 313 tiles * 16
#define N_TILES  313
#define JT_PER_WG 8              // j-tiles per wave in K2
#define N_CLS    80
#define OUT_COLS 84
#define ROW_W32  160             // mask row stride in u32 words (157 used)
#define ROW_W16  320             // same row stride in u16 units
#define ROW_V4   40              // same row stride in uint4 units
#define SORT_N   8192
#define SCORE_THR 0.05f
#define NMS_THR   0.5f
#define MAX_KEEP  300

typedef __attribute__((ext_vector_type(2))) float v2f;
typedef __attribute__((ext_vector_type(8))) float v8f;

// ---------------------------------------------------------------- kernel 1 --
__global__ __launch_bounds__(256) void nms_prep_kernel(
    const float* __restrict__ boxes,
    float* __restrict__ xlo, float* __restrict__ ylo,
    float* __restrict__ xhi, float* __restrict__ yhi,
    float* __restrict__ area,
    unsigned* __restrict__ maskWords, int maskWordCount,
    float* __restrict__ out)
{
    int idx    = blockIdx.x * blockDim.x + threadIdx.x;
    int stride = gridDim.x * blockDim.x;

    for (int i = idx; i < N_PAD; i += stride) {
        float x1, y1, x2, y2, ar;
        if (i < N_ANCH) {
            x1 = boxes[i * 4 + 0];
            y1 = boxes[i * 4 + 1];
            x2 = boxes[i * 4 + 2];
            y2 = boxes[i * 4 + 3];
            ar = (x2 - x1) * (y2 - y1);
            out[i * OUT_COLS + 0] = x1;
            out[i * OUT_COLS + 1] = y1;
            out[i * OUT_COLS + 2] = x2;
            out[i * OUT_COLS + 3] = y2;
        } else {
            // degenerate far-away box: zero area, zero intersection with all
            x1 = y1 = x2 = y2 = 1.0e8f;
            ar = 0.0f;
        }
        xlo[i] = x1; ylo[i] = y1; xhi[i] = x2; yhi[i] = y2; area[i] = ar;
    }
    for (int w = idx; w < maskWordCount; w += stride)
        maskWords[w] = 0u;
}

// ---------------------------------------------------------------- kernel 2 --
// One wave handles one i-tile x 8 j-tiles. The A fragment and i-tile coords
// are loaded once; each j-tile issues one WMMA for the area_i+area_j term.
__global__ __launch_bounds__(32) void nms_iou_mask_kernel(
    const float* __restrict__ xlo, const float* __restrict__ ylo,
    const float* __restrict__ xhi, const float* __restrict__ yhi,
    const float* __restrict__ area,
    unsigned short* __restrict__ mask16)
{
    const int jtBase = blockIdx.x * JT_PER_WG;
    const int it     = blockIdx.y;
    const int lane   = threadIdx.x;        // 0..31, wave32
    const int n      = lane & 15;
    const int loHalf = (lane < 16);
    const int mhi    = (lane >> 4) << 3;
    const int rowBase = it * 16;

    __shared__ float six1[16], siy1[16], six2[16], siy2[16];

    // ---- i-tile: LDS stage + A fragment (once per block) ----
    const int gi0 = rowBase + n;
    float ix1 = xlo[gi0], iy1 = ylo[gi0], ix2 = xhi[gi0], iy2 = yhi[gi0];
    float ai  = area[gi0];
    if (lane < 16) {
        six1[lane] = ix1; siy1[lane] = iy1;
        six2[lane] = ix2; siy2[lane] = iy2;
    }
    v2f a;
    a.x = loHalf ? ai   : 0.0f;   // K=0 column: area_i (lanes 16-31 pad K=2)
    a.y = loHalf ? 1.0f : 0.0f;   // K=1 column: ones   (lanes 16-31 pad K=3)
    __syncthreads();

    #pragma unroll
    for (int q = 0; q < JT_PER_WG; ++q) {
        int jt = jtBase + q;
        if (jt >= N_TILES) break;           // uniform across the wave

        // ---- j-tile straight to registers (half-wave broadcast loads) ----
        int gj = jt * 16 + n;
        float jx1 = xlo[gj], jy1 = ylo[gj], jx2 = xhi[gj], jy2 = yhi[gj];
        float aj  = area[gj];
        v2f b;
        b.x = loHalf ? 1.0f : 0.0f;         // K=0 row of ones
        b.y = loHalf ? aj   : 0.0f;         // K=1 row = area_j

        v8f c = {};
        // d[r] @ lane = area_i[m] + area_j[n], m = r + mhi
        v8f d = __builtin_amdgcn_wmma_f32_16x16x4_f32(
            false, a, false, b, (short)0, c, false, false);

        #pragma unroll
        for (int r = 0; r < 8; ++r) {
            int   m  = r + mhi;
            float iw = fminf(six2[m], jx2) - fmaxf(six1[m], jx1);
            float ih = fminf(siy2[m], jy2) - fmaxf(siy1[m], jy1);
            iw = fmaxf(iw, 0.0f);
            ih = fmaxf(ih, 0.0f);
            float inter = iw * ih;
            float uni   = fmaxf(d[r] - inter, 1e-8f);  // area_i+area_j-inter
            int   gi    = rowBase + m;
            // iou > 0.5  <=>  inter > 0.5*uni  (0.5*uni is exact; uni > 0)
            bool  p     = (inter > NMS_THR * uni) && (gi != gj);

            unsigned bal = __builtin_amdgcn_ballot_w32(p);
            // bits 0..15  -> row (rowBase+r),   cols jt*16..+15
            // bits 16..31 -> row (rowBase+r+8), cols jt*16..+15
            if (lane == 0)
                mask16[(unsigned)(rowBase + r)     * ROW_W16 + jt] =
                    (unsigned short)(bal & 0xFFFFu);
            if (lane == 16)
                mask16[(unsigned)(rowBase + r + 8) * ROW_W16 + jt] =
                    (unsigned short)(bal >> 16);
        }
    }
}

// ---------------------------------------------------------------- kernel 3 --
__device__ __forceinline__ bool ord_before(float sa, int a, float sb, int b) {
    // descending score, ascending index (matches stable argsort of -scores)
    return (sa > sb) || (sa == sb && a < b);
}

__global__ __launch_bounds__(256) void nms_scan_kernel(
    const float* __restrict__ cls,
    const unsigned* __restrict__ mask,
    float* __restrict__ out)
{
    const int c   = blockIdx.x;       // class
    const int tid = threadIdx.x;

    __shared__ float    sc[N_PAD];
    __shared__ int      ord[SORT_N];
    __shared__ uint4    sup4[ROW_V4];
    __shared__ uint4    kept4[ROW_V4];
    __shared__ int      aliveOi;
    __shared__ int      count;

    unsigned* sup  = (unsigned*)sup4;
    unsigned* kept = (unsigned*)kept4;

    for (int i = tid; i < N_PAD; i += 256)
        sc[i] = (i < N_ANCH) ? cls[i * N_CLS + c] : -1.0f;
    for (int t = tid; t < SORT_N; t += 256)
        ord[t] = t;
    if (tid < ROW_V4) {
        sup4[tid]  = make_uint4(0u, 0u, 0u, 0u);
        kept4[tid] = make_uint4(0u, 0u, 0u, 0u);
    }
    if (tid == 0) count = 0;
    __syncthreads();

    // ---- bitonic argsort over SORT_N entries (sentinels sort last) ----
    for (int k = 2; k <= SORT_N; k <<= 1) {
        for (int j = k >> 1; j > 0; j >>= 1) {
            for (int i = tid; i < SORT_N; i += 256) {
                int ixj = i ^ j;
                if (ixj > i) {
                    int va = ord[i], vb = ord[ixj];
                    float sa = (va < N_ANCH) ? sc[va] : -1.0e30f;
                    float sb = (vb < N_ANCH) ? sc[vb] : -1.0e30f;
                    bool up = (i & k) == 0;
                    bool sw = up ? ord_before(sb, vb, sa, va)
                                 : ord_before(sa, va, sb, vb);
                    if (sw) { ord[i] = vb; ord[ixj] = va; }
                }
            }
            __syncthreads();
        }
    }

    // ---- sequential greedy scan over sorted order ----
    for (int t = 0; t < N_ANCH; ++t) {
        if (tid == 0) {
            int  oi      = ord[t];
            bool is_sup  = (sup[oi >> 5] >> (oi & 31)) & 1u;
            bool alive   = !is_sup && (sc[oi] > SCORE_THR) && (count < MAX_KEEP);
            if (alive) {
                ++count;
                kept[oi >> 5] |= 1u << (oi & 31);
                aliveOi = oi;
            } else {
                aliveOi = -1;
            }
        }
        if (tid == 64 && t + 1 < N_ANCH) {
            // warm L2->WGP path for the next candidate's mask row
            __builtin_prefetch(&mask[(unsigned)ord[t + 1] * ROW_W32], 0, 0);
        }
        __syncthreads();
        int oi = aliveOi;
        if (oi >= 0 && tid < ROW_V4) {
            // row stride is 640B -> uint4 aligned; pad words are zeroed
            const uint4* row = (const uint4*)(mask + (unsigned)oi * ROW_W32);
            uint4 rv = row[tid];
            uint4 sv = sup4[tid];
            sv.x |= rv.x; sv.y |= rv.y; sv.z |= rv.z; sv.w |= rv.w;
            sup4[tid] = sv;
        }
        __syncthreads();
    }

    // ---- emit this class's score column ----
    for (int i = tid; i < N_ANCH; i += 256) {
        bool k = (kept[i >> 5] >> (i & 31)) & 1u;
        out[i * OUT_COLS + 4 + c] = k ? sc[i] : 0.0f;
    }
}

// ------------------------------------------------------------------- launch --
extern "C" void kernel_launch(void* const* d_in, const int* in_sizes, int n_in,
                              void* d_out, int out_size, void* d_ws, size_t ws_size,
                              hipStream_t stream) {
    const float* boxes = (const float*)d_in[0];   // (1,5000,4)
    const float* cls   = (const float*)d_in[1];   // (1,5000,80)
    float*       out   = (float*)d_out;           // (1,5000,84)

    char* ws = (char*)d_ws;
    float* xlo  = (float*)ws;
    float* ylo  = xlo + N_PAD;
    float* xhi  = ylo + N_PAD;
    float* yhi  = xhi + N_PAD;
    float* area = yhi + N_PAD;
    size_t boxBytes = (size_t)5 * N_PAD * sizeof(float);          // 100160
    size_t maskOff  = (boxBytes + 255) & ~(size_t)255;            // 100352
    unsigned*       mask32 = (unsigned*)(ws + maskOff);
    unsigned short* mask16 = (unsigned short*)mask32;
    int maskWordCount = N_PAD * ROW_W32;                          // 801280 u32

    nms_prep_kernel<<<1024, 256, 0, stream>>>(
        boxes, xlo, ylo, xhi, yhi, area, mask32, maskWordCount, out);

    dim3 grid2((N_TILES + JT_PER_WG - 1) / JT_PER_WG, N_TILES);   // 40 x 313
    nms_iou_mask_kernel<<<grid2, 32, 0, stream>>>(
        xlo, ylo, xhi, yhi, area, mask16);

    nms_scan_kernel<<<N_CLS, 256, 0, stream>>>(cls, mask32, out);
}